// FRFSIMLoss_74294344286652
// MI455X (gfx1250) — compile-verified
//
#include <hip/hip_runtime.h>
#include <math.h>

typedef float v2f __attribute__((ext_vector_type(2)));
typedef float v8f __attribute__((ext_vector_type(8)));

#define EPSF    1e-8f
#define IMG_W   512
#define IMG_HW  (512 * 512)

// 1D Gaussian, ks=7, sigma=7/6, normalized (matches _gauss_kernel)
#define G0 0.0125602f
#define G1 0.0788278f
#define G2 0.2372966f
#define G3 0.3426314f

// ---------- monotonic float<->uint keys for atomic min/max ----------
__device__ __forceinline__ unsigned fkey(float f) {
  unsigned u = __float_as_uint(f);
  return (u & 0x80000000u) ? ~u : (u | 0x80000000u);
}
__device__ __forceinline__ float funkey(unsigned k) {
  unsigned u = (k & 0x80000000u) ? (k & 0x7fffffffu) : ~k;
  return __uint_as_float(u);
}

// base-2 log/exp mapping to v_log_f32 / v_exp_f32
__device__ __forceinline__ float dev_log2(float x) {
#if __has_builtin(__builtin_amdgcn_logf)
  return __builtin_amdgcn_logf(x);
#else
  return log2f(x);
#endif
}
__device__ __forceinline__ float dev_exp2(float x) {
#if __has_builtin(__builtin_amdgcn_exp2f)
  return __builtin_amdgcn_exp2f(x);
#else
  return exp2f(x);
#endif
}
__device__ __forceinline__ float fpow_pos(float x, float b) {
  // x^b for x >= 0 (clamped away from 0 so log is finite)
  return dev_exp2(b * dev_log2(fmaxf(x, 1e-30f)));
}

struct KC { v2f k[6]; };  // banded-Gaussian WMMA fragment, per lane

// =====================================================================
// k0: zero/initialize accumulators (graph-safe, runs every launch)
// =====================================================================
__global__ void k0_init(float* out, unsigned* minkey, unsigned* maxkey,
                        float* sdsum, int B) {
  int t = threadIdx.x;
  if (t == 0) out[0] = 0.f;
  if (t < B) { minkey[t] = 0xFFFFFFFFu; maxkey[t] = 0u; sdsum[t] = 0.f; }
}

// =====================================================================
// k1: grayscale + per-sample min/max of Dg.  One block never crosses a
// sample boundary (HW % 256 == 0).
// =====================================================================
__global__ __launch_bounds__(256) void k1_gray(
    const float* __restrict__ D, const float* __restrict__ R,
    float* __restrict__ Dg, float* __restrict__ Rg,
    unsigned* __restrict__ minkey, unsigned* __restrict__ maxkey) {
  int idx = blockIdx.x * 256 + threadIdx.x;
  int b = idx / IMG_HW;
  int p = idx - b * IMG_HW;
  const float* Dp = D + (size_t)b * 3 * IMG_HW + p;
  const float* Rp = R + (size_t)b * 3 * IMG_HW + p;
  float dg = 0.2989f * Dp[0] + 0.587f * Dp[IMG_HW] + 0.114f * Dp[2 * IMG_HW];
  float rg = 0.2989f * Rp[0] + 0.587f * Rp[IMG_HW] + 0.114f * Rp[2 * IMG_HW];
  Dg[idx] = dg;
  Rg[idx] = rg;

  __shared__ float smn[256], smx[256];
  smn[threadIdx.x] = dg;
  smx[threadIdx.x] = dg;
  __syncthreads();
  for (int s = 128; s >= 1; s >>= 1) {
    if (threadIdx.x < (unsigned)s) {
      smn[threadIdx.x] = fminf(smn[threadIdx.x], smn[threadIdx.x + s]);
      smx[threadIdx.x] = fmaxf(smx[threadIdx.x], smx[threadIdx.x + s]);
    }
    __syncthreads();
  }
  if (threadIdx.x == 0) {
    atomicMin(&minkey[b], fkey(smn[0]));
    atomicMax(&maxkey[b], fkey(smx[0]));
  }
}

// =====================================================================
// k2: per-sample SSIM-like constants from dynamic range d
// consts layout per sample: [CM, CD, CG, CC, b1, b2]
// =====================================================================
__global__ void k2_consts(const unsigned* minkey, const unsigned* maxkey,
                          float* consts, int B) {
  int b = threadIdx.x;
  if (b < B) {
    float d  = funkey(maxkey[b]) - funkey(minkey[b]);
    float CM = (0.01f * d) * (0.01f * d);
    float CG = (0.03f * d) * (0.03f * d);
    consts[b * 6 + 0] = CM;
    consts[b * 6 + 1] = CM * 0.5f;  // CD
    consts[b * 6 + 2] = CG;
    consts[b * 6 + 3] = CG * 0.5f;  // CC
  }
}

// =====================================================================
// k3: per-sample sum of dark-channel similarity SD
// =====================================================================
__global__ __launch_bounds__(256) void k3_sd(
    const float* __restrict__ D, const float* __restrict__ R,
    const float* __restrict__ consts, float* __restrict__ sdsum) {
  int idx = blockIdx.x * 256 + threadIdx.x;
  int b = idx / IMG_HW;
  int p = idx - b * IMG_HW;
  const float* Dp = D + (size_t)b * 3 * IMG_HW + p;
  const float* Rp = R + (size_t)b * 3 * IMG_HW + p;
  float CD = consts[b * 6 + 1];
  float dmn = fminf(Dp[0], fminf(Dp[IMG_HW], Dp[2 * IMG_HW]));
  float rmn = fminf(Rp[0], fminf(Rp[IMG_HW], Rp[2 * IMG_HW]));
  float a = dmn * (1.f / 255.f), c = rmn * (1.f / 255.f);
  float SD = (2.f * a * c + CD) / (a * a + c * c + CD + EPSF);

  __shared__ float ssum[256];
  ssum[threadIdx.x] = SD;
  __syncthreads();
  for (int s = 128; s >= 1; s >>= 1) {
    if (threadIdx.x < (unsigned)s)
      ssum[threadIdx.x] += ssum[threadIdx.x + s];
    __syncthreads();
  }
  if (threadIdx.x == 0) atomicAdd(&sdsum[b], ssum[0]);
}

// =====================================================================
// k4: adaptive exponents b1, b2 from mean_SD
// =====================================================================
__global__ void k4_exponents(const float* sdsum, float* consts, int B) {
  int b = threadIdx.x;
  if (b < B) {
    float m = sdsum[b] * (1.f / (float)IMG_HW);
    float b1, b2;
    if (m > 0.85f && m <= 1.0f)      { b1 = 0.2f; b2 = 0.8f; }
    else if (m >= 0.f && m <= 0.85f) { b1 = 0.8f; b2 = 0.2f; }
    else                             { b1 = 0.5f; b2 = 0.5f; }
    consts[b * 6 + 4] = b1;
    consts[b * 6 + 5] = b2;
  }
}

// =====================================================================
// Wave-level separable 7x7 Gaussian conv of one 16x16 output tile via
// chained V_WMMA_F32_16X16X4_F32.
//   stage 1: Mid(22x16) = In(22x22) * Kh(22x16), Kh[c][n] = g[c-n]
//   stage 2: Out(16x16) = V(16x22) * Mid(22x16), V[m][r]  = g[r-m]
// Per-lane fragment value for both Kh (B-operand, stage 1) and V
// (A-operand, stage 2) is g[K - (lane&15)], K = 4t + v + 2*(lane>>4).
// A-frag layout: a[v] = A[lane&15][4t+v+2*(lane>>4)]
// B-frag layout: b[v] = B[4t+v+2*(lane>>4)][lane&15]
// D-frag layout: d[r] = D[r + 8*(lane>>4)][lane&15]
//
// mid has 32 rows: the rb=1 block stores all 8 result rows
// unconditionally (rows 22..31 are exact zeros because the
// corresponding A rows were zeroed), so stage-2 reads of K=22,23 need
// no predication.  Stage-1 A loads use clamped addresses + value
// select instead of exec-predicated loads.
// =====================================================================
__device__ __forceinline__ v8f wave_conv7(const float (*tile)[24],
                                          float (*mid)[17], const KC& kc,
                                          int n, int h, bool sq) {
  // ---- stage 1: horizontal (rows 0-15, then rows 16-21 padded) ----
#pragma unroll
  for (int rb = 0; rb < 2; ++rb) {
    v8f acc = {0.f, 0.f, 0.f, 0.f, 0.f, 0.f, 0.f, 0.f};
    int arow = n + 16 * rb;             // A-matrix row = input row
    int rowc = arow < 22 ? arow : 21;   // clamped LDS row address
#pragma unroll
    for (int t = 0; t < 6; ++t) {
      v2f a;
#pragma unroll
      for (int v = 0; v < 2; ++v) {
        int K  = 4 * t + v + 2 * h;     // input column index
        int Kc = K < 22 ? K : 21;       // clamped LDS column address
        float x = tile[rowc][Kc];       // unconditional ds_load
        x = (arow < 22 && K < 22) ? x : 0.f;  // value select (cndmask)
        a[v] = sq ? x * x : x;
      }
      acc = __builtin_amdgcn_wmma_f32_16x16x4_f32(
          false, a, false, kc.k[t], (short)0, acc, false, false);
    }
#pragma unroll
    for (int r = 0; r < 8; ++r) {
      int row = r + 8 * h + 16 * rb;    // 0..31: always in range (32 rows)
      mid[row][n] = acc[r];             // rows >=22 are exact zeros
    }
  }
  __syncthreads();
  // ---- stage 2: vertical (unconditional: rows 22,23 hold zeros) ----
  v8f outv = {0.f, 0.f, 0.f, 0.f, 0.f, 0.f, 0.f, 0.f};
#pragma unroll
  for (int t = 0; t < 6; ++t) {
    v2f bf;
#pragma unroll
    for (int v = 0; v < 2; ++v) {
      int K = 4 * t + v + 2 * h;        // Mid row index, <= 23 < 32
      bf[v] = mid[K][n];
    }
    outv = __builtin_amdgcn_wmma_f32_16x16x4_f32(
        false, kc.k[t], false, bf, (short)0, outv, false, false);
  }
  __syncthreads();  // mid free for next field
  return outv;
}

// =====================================================================
// k5: fused MSCN + Sobel + similarity maps + final mean.
// 256 threads = 8 waves; each wave owns one 16x16 output tile and a
// private LDS slice (22x22 zero-padded Dg/Rg tile + 32x16 conv scratch).
// =====================================================================
__global__ __launch_bounds__(256) void k5_main(
    const float* __restrict__ D, const float* __restrict__ R,
    const float* __restrict__ Dg, const float* __restrict__ Rg,
    const float* __restrict__ consts, float* __restrict__ out, float scale) {
  __shared__ float sD[8][22][24];
  __shared__ float sR[8][22][24];
  __shared__ float sM[8][32][17];

  const int lane = threadIdx.x & 31;
  const int wv   = threadIdx.x >> 5;
  const int n    = lane & 15;
  const int h    = lane >> 4;

  int tileId = blockIdx.x * 8 + wv;
  int b  = tileId >> 10;          // 1024 tiles per sample (32x32)
  int t  = tileId & 1023;
  int y0 = (t >> 5) << 4;
  int x0 = (t & 31) << 4;

  const float CM = consts[b * 6 + 0];
  const float CD = consts[b * 6 + 1];
  const float CG = consts[b * 6 + 2];
  const float CC = consts[b * 6 + 3];
  const float b1 = consts[b * 6 + 4];
  const float b2 = consts[b * 6 + 5];

  // banded-Gaussian constant fragments
  const float g7[7] = {G0, G1, G2, G3, G2, G1, G0};
  KC kc;
#pragma unroll
  for (int tt = 0; tt < 6; ++tt)
#pragma unroll
    for (int v = 0; v < 2; ++v) {
      int K = 4 * tt + v + 2 * h;
      int d = K - n;
      kc.k[tt][v] = (K < 22 && d >= 0 && d <= 6) ? g7[d] : 0.f;
    }

  // load zero-padded 22x22 gray tiles into this wave's LDS slice
  const float* dg = Dg + (size_t)b * IMG_HW;
  const float* rg = Rg + (size_t)b * IMG_HW;
  for (int i = lane; i < 22 * 22; i += 32) {
    int r = i / 22, c = i - r * 22;
    int y = y0 + r - 3, x = x0 + c - 3;
    bool ok = (y >= 0 && y < IMG_W && x >= 0 && x < IMG_W);
    int off = y * IMG_W + x;
    sD[wv][r][c] = ok ? dg[off] : 0.f;
    sR[wv][r][c] = ok ? rg[off] : 0.f;
  }
  __syncthreads();

  v8f muD = wave_conv7(sD[wv], sM[wv], kc, n, h, false);
  v8f e2D = wave_conv7(sD[wv], sM[wv], kc, n, h, true);
  v8f muR = wave_conv7(sR[wv], sM[wv], kc, n, h, false);
  v8f e2R = wave_conv7(sR[wv], sM[wv], kc, n, h, true);

  const float* Dp = D + (size_t)b * 3 * IMG_HW;
  const float* Rp = R + (size_t)b * 3 * IMG_HW;

  float lsum = 0.f;
#pragma unroll
  for (int r = 0; r < 8; ++r) {
    int m = r + 8 * h;          // row within tile (D-frag layout)
    int y = y0 + m, x = x0 + n;

    // ----- MSCN structure similarity -----
    float gD = sD[wv][3 + m][3 + n];
    float gR = sR[wv][3 + m][3 + n];
    float mud = muD[r], mur = muR[r];
    float sigD = sqrtf(fabsf(e2D[r] - mud * mud + EPSF));
    float sigR = sqrtf(fabsf(e2R[r] - mur * mur + EPSF));
    float Dm = (gD - mud) / (sigD + EPSF);
    float Rm = (gR - mur) / (sigR + EPSF);
    float SM = (2.f * Dm * Rm + CM) / (Dm * Dm + Rm * Rm + CM);

    // ----- Sobel gradient similarity (zero-padded, cross-correlation) -----
    float a00 = sD[wv][2 + m][2 + n], a01 = sD[wv][2 + m][3 + n], a02 = sD[wv][2 + m][4 + n];
    float a10 = sD[wv][3 + m][2 + n],                              a12 = sD[wv][3 + m][4 + n];
    float a20 = sD[wv][4 + m][2 + n], a21 = sD[wv][4 + m][3 + n], a22 = sD[wv][4 + m][4 + n];
    float gxD = (a00 - a02) + 2.f * (a10 - a12) + (a20 - a22);
    float gyD = (a00 + 2.f * a01 + a02) - (a20 + 2.f * a21 + a22);
    float Dgr = sqrtf(gxD * gxD + gyD * gyD + EPSF);

    float c00 = sR[wv][2 + m][2 + n], c01 = sR[wv][2 + m][3 + n], c02 = sR[wv][2 + m][4 + n];
    float c10 = sR[wv][3 + m][2 + n],                              c12 = sR[wv][3 + m][4 + n];
    float c20 = sR[wv][4 + m][2 + n], c21 = sR[wv][4 + m][3 + n], c22 = sR[wv][4 + m][4 + n];
    float gxR = (c00 - c02) + 2.f * (c10 - c12) + (c20 - c22);
    float gyR = (c00 + 2.f * c01 + c02) - (c20 + 2.f * c21 + c22);
    float Rgr = sqrtf(gxR * gxR + gyR * gyR + EPSF);

    float SG = (2.f * Dgr * Rgr + CG) / (Dgr * Dgr + Rgr * Rgr + CG + EPSF);

    // ----- dark channel + HSV color similarities (from D, R directly) -----
    int po = y * IMG_W + x;
    float d0 = Dp[po], d1 = Dp[IMG_HW + po], d2 = Dp[2 * IMG_HW + po];
    float r0 = Rp[po], r1 = Rp[IMG_HW + po], r2 = Rp[2 * IMG_HW + po];
    float dmn = fminf(d0, fminf(d1, d2)), dmx = fmaxf(d0, fmaxf(d1, d2));
    float rmn = fminf(r0, fminf(r1, r2)), rmx = fmaxf(r0, fmaxf(r1, r2));
    float Ddc = dmn * (1.f / 255.f), Rdc = rmn * (1.f / 255.f);
    float SDv = (2.f * Ddc * Rdc + CD) / (Ddc * Ddc + Rdc * Rdc + CD + EPSF);
    float Dc = ((dmx - dmn + EPSF) / (dmx + EPSF)) * dmx;
    float Rc = ((rmx - rmn + EPSF) / (rmx + EPSF)) * rmx;
    float SC = (2.f * Dc * Rc + CC) / (Dc * Dc + Rc * Rc + CC + EPSF);

    // ----- adaptive-exponent fusion -----
    float p1 = fpow_pos(fabsf(SM * SDv), b1);
    float p2 = fpow_pos(fmaxf(SG * SC, 0.f), b2);
    lsum += fabsf(p1 * p2);
  }

  // wave reduce + single atomic per wave
#pragma unroll
  for (int s = 16; s >= 1; s >>= 1) lsum += __shfl_xor(lsum, s, 32);
  if (lane == 0) atomicAdd(out, lsum * scale);
}

// =====================================================================
// launch
// =====================================================================
extern "C" void kernel_launch(void* const* d_in, const int* in_sizes, int n_in,
                              void* d_out, int out_size, void* d_ws, size_t ws_size,
                              hipStream_t stream) {
  (void)n_in; (void)out_size; (void)ws_size;
  const float* D = (const float*)d_in[0];
  const float* R = (const float*)d_in[1];
  float* out = (float*)d_out;

  const int B = in_sizes[0] / (3 * IMG_HW);  // 32

  // workspace: Dg | Rg | minkey[B] | maxkey[B] | sdsum[B] | consts[B*6]
  float* wsDg = (float*)d_ws;
  float* wsRg = wsDg + (size_t)B * IMG_HW;
  unsigned* minkey = (unsigned*)(wsRg + (size_t)B * IMG_HW);
  unsigned* maxkey = minkey + B;
  float* sdsum = (float*)(maxkey + B);
  float* consts = sdsum + B;

  const int pixBlocks = (B * IMG_HW) / 256;   // 32768, sample-aligned
  const int tileBlocks = (B * 1024) / 8;      // 4096 (8 waves/block)
  const float scale = 1.0f / ((float)B * (float)IMG_HW);

  k0_init<<<1, 256, 0, stream>>>(out, minkey, maxkey, sdsum, B);
  k1_gray<<<pixBlocks, 256, 0, stream>>>(D, R, wsDg, wsRg, minkey, maxkey);
  k2_consts<<<1, 64, 0, stream>>>(minkey, maxkey, consts, B);
  k3_sd<<<pixBlocks, 256, 0, stream>>>(D, R, consts, sdsum);
  k4_exponents<<<1, 64, 0, stream>>>(sdsum, consts, B);
  k5_main<<<tileBlocks, 256, 0, stream>>>(D, R, wsDg, wsRg, consts, out, scale);
}